// NMS_FPN_56427280335586
// MI455X (gfx1250) — compile-verified
//
#include <hip/hip_runtime.h>
#include <stdint.h>

// ---------------- problem constants (match reference) ----------------
#define NUM_CLASSES   80
#define MAX_PER_CLASS 200           // BATCH_SIZE * 100
#define IOU_THR       0.5f

#define BLOCK 256                   // 8 wave32s
#define CAP   1024                  // max candidates kept per class (mean ~125, std ~11)
#define CHUNK 512                   // staging chunk (elements)
#define ITER_PER_THREAD   (CHUNK / BLOCK)        // 2
#define ISSUES_PER_THREAD (2 * ITER_PER_THREAD)  // 4 async b32 per thread per chunk

// ---------------- gfx1250 async global->LDS path ----------------
#if defined(__has_builtin)
#if __has_builtin(__builtin_amdgcn_global_load_async_to_lds_b32) && \
    __has_builtin(__builtin_amdgcn_s_wait_asynccnt)
#define USE_ASYNC 1
#endif
#endif
#ifndef USE_ASYNC
#define USE_ASYNC 0
#endif

#if USE_ASYNC
typedef __attribute__((address_space(1))) int GInt;   // global (as1) int
typedef __attribute__((address_space(3))) int LInt;   // LDS (as3) int
static __device__ __forceinline__ GInt* as_global_i32(const void* p) {
  return (GInt*)(unsigned long long)(uintptr_t)p;
}
static __device__ __forceinline__ LInt* as_lds_i32(void* p) {
  // generic LDS address: low 32 bits are the LDS offset (AMDGPU flat->local lowering)
  return (LInt*)(unsigned int)(uintptr_t)p;
}
#endif

__global__ __launch_bounds__(BLOCK)
void NMS_FPN_56427280335586_kernel(const int*   __restrict__ anchor_idxes, // [N,2]
                                   const float* __restrict__ anchors,      // [N,4]
                                   const int*   __restrict__ class_ids,    // [N]
                                   const float* __restrict__ scores,       // [N]
                                   float*       __restrict__ out,          // 128000 floats
                                   int n) {
  const int c   = (int)blockIdx.x;     // one workgroup per class
  const int tid = (int)threadIdx.x;
  const int CM  = NUM_CLASSES * MAX_PER_CLASS; // 16000

  // candidate state (SoA, LDS-resident for the whole greedy loop)
  __shared__ float s_y1[CAP], s_x1[CAP], s_y2[CAP], s_x2[CAP], s_area[CAP];
  __shared__ float s_score[CAP];
  __shared__ int   s_gidx[CAP];
  __shared__ unsigned char s_active[CAP];
  __shared__ int   s_count;
#if USE_ASYNC
  __shared__ int   s_cls[2][CHUNK];
  __shared__ float s_sc[2][CHUNK];
#endif
  __shared__ float s_ws[BLOCK / 32];
  __shared__ int   s_wi[BLOCK / 32];
  __shared__ int   s_wslot[BLOCK / 32];
  __shared__ int   s_sel;
  __shared__ float s_selv[5];          // y1,x1,y2,x2,area of selected box

  if (tid == 0) s_count = 0;
  __syncthreads();

  // ---------------- phase 1: gather this class's boxes into LDS ----------------
#if USE_ASYNC
  // prologue: issue chunk 0 into buffer 0 (addresses clamped -> uniform issue count)
  {
#pragma unroll
    for (int k = 0; k < ITER_PER_THREAD; ++k) {
      int i = tid + k * BLOCK;
      int g = i; if (g >= n) g = n - 1;
      __builtin_amdgcn_global_load_async_to_lds_b32(as_global_i32(class_ids + g),
                                                    as_lds_i32(&s_cls[0][i]), 0, 0);
      __builtin_amdgcn_global_load_async_to_lds_b32(as_global_i32(scores + g),
                                                    as_lds_i32(&s_sc[0][i]), 0, 0);
    }
  }
  int buf = 0;
  for (int cb = 0; cb < n; cb += CHUNK) {
    const bool has_next = (cb + CHUNK) < n;
    if (has_next) {
      const int base = cb + CHUNK, nb = buf ^ 1;
#pragma unroll
      for (int k = 0; k < ITER_PER_THREAD; ++k) {
        int i = tid + k * BLOCK;
        int g = base + i; if (g >= n) g = n - 1;
        __builtin_amdgcn_global_load_async_to_lds_b32(as_global_i32(class_ids + g),
                                                      as_lds_i32(&s_cls[nb][i]), 0, 0);
        __builtin_amdgcn_global_load_async_to_lds_b32(as_global_i32(scores + g),
                                                      as_lds_i32(&s_sc[nb][i]), 0, 0);
      }
      __builtin_amdgcn_s_wait_asynccnt(ISSUES_PER_THREAD); // current chunk complete
    } else {
      __builtin_amdgcn_s_wait_asynccnt(0);
    }
    // each thread consumes exactly the elements it issued -> no cross-wave hazard
#pragma unroll
    for (int k = 0; k < ITER_PER_THREAD; ++k) {
      int i = tid + k * BLOCK;
      int g = cb + i;
      if (g < n && s_cls[buf][i] == c) {
        int pos = atomicAdd(&s_count, 1);
        if (pos < CAP) {
          float4 b = reinterpret_cast<const float4*>(anchors)[g]; // (y1,x1,y2,x2)
          s_y1[pos] = b.x; s_x1[pos] = b.y; s_y2[pos] = b.z; s_x2[pos] = b.w;
          s_area[pos]  = (b.z - b.x) * (b.w - b.y);
          s_score[pos] = s_sc[buf][i];
          s_gidx[pos]  = g;
          s_active[pos] = 1;
        }
      }
    }
    buf ^= 1;
  }
#else
  for (int g = tid; g < n; g += BLOCK) {
    __builtin_prefetch(class_ids + g + 4 * BLOCK, 0, 0); // global_prefetch_b8
    if (class_ids[g] == c) {
      int pos = atomicAdd(&s_count, 1);
      if (pos < CAP) {
        float4 b = reinterpret_cast<const float4*>(anchors)[g];
        s_y1[pos] = b.x; s_x1[pos] = b.y; s_y2[pos] = b.z; s_x2[pos] = b.w;
        s_area[pos]  = (b.z - b.x) * (b.w - b.y);
        s_score[pos] = scores[g];
        s_gidx[pos]  = g;
        s_active[pos] = 1;
      }
    }
  }
#endif
  __syncthreads();
  const int count = (s_count < CAP) ? s_count : CAP;

  // ---------------- phase 2: 200 greedy NMS steps, entirely in LDS ----------------
  for (int step = 0; step < MAX_PER_CLASS; ++step) {
    // local argmax over this thread's slots; tie-break = smallest global index
    float best_s = -1.0f; int best_i = 0x7FFFFFFF; int best_slot = -1;
    for (int j = tid; j < count; j += BLOCK) {
      if (s_active[j]) {
        float sc = s_score[j]; int gi = s_gidx[j];
        if (sc > best_s || (sc == best_s && gi < best_i)) {
          best_s = sc; best_i = gi; best_slot = j;
        }
      }
    }
    // wave32 reduction
#pragma unroll
    for (int off = 16; off > 0; off >>= 1) {
      float os  = __shfl_xor(best_s, off, 32);
      int   oi  = __shfl_xor(best_i, off, 32);
      int   osl = __shfl_xor(best_slot, off, 32);
      if (os > best_s || (os == best_s && oi < best_i)) {
        best_s = os; best_i = oi; best_slot = osl;
      }
    }
    if ((tid & 31) == 0) {
      int w = tid >> 5;
      s_ws[w] = best_s; s_wi[w] = best_i; s_wslot[w] = best_slot;
    }
    __syncthreads();                                   // B1
    if (tid == 0) {
#pragma unroll
      for (int w = 1; w < BLOCK / 32; ++w) {
        if (s_ws[w] > best_s || (s_ws[w] == best_s && s_wi[w] < best_i)) {
          best_s = s_ws[w]; best_i = s_wi[w]; best_slot = s_wslot[w];
        }
      }
      s_sel = best_slot;
      const int slot = c * MAX_PER_CLASS + step;
      float* oa = out + 4 * slot;
      if (best_slot >= 0) {
        float y1 = s_y1[best_slot], x1 = s_x1[best_slot];
        float y2 = s_y2[best_slot], x2 = s_x2[best_slot];
        s_selv[0] = y1; s_selv[1] = x1; s_selv[2] = y2; s_selv[3] = x2;
        s_selv[4] = s_area[best_slot];
        oa[0] = y1; oa[1] = x1; oa[2] = y2; oa[3] = x2;
        out[4 * CM + slot] = (float)c;                               // class_sel
        out[5 * CM + slot] = s_score[best_slot];                     // scores_sel
        out[6 * CM + slot] = (float)anchor_idxes[2 * s_gidx[best_slot]]; // batch idx
        out[7 * CM + slot] = 1.0f;                                   // valid
      } else {
        oa[0] = 0.0f; oa[1] = 0.0f; oa[2] = 0.0f; oa[3] = 0.0f;
        out[4 * CM + slot] = -1.0f;
        out[5 * CM + slot] =  0.0f;
        out[6 * CM + slot] = -1.0f;
        out[7 * CM + slot] =  0.0f;
      }
    }
    __syncthreads();                                   // B2
    const int sel = s_sel;
    if (sel >= 0) {
      const float by1 = s_selv[0], bx1 = s_selv[1];
      const float by2 = s_selv[2], bx2 = s_selv[3], ba = s_selv[4];
      for (int j = tid; j < count; j += BLOCK) {
        if (s_active[j]) {
          float iy1 = fmaxf(s_y1[j], by1);
          float ix1 = fmaxf(s_x1[j], bx1);
          float iy2 = fminf(s_y2[j], by2);
          float ix2 = fminf(s_x2[j], bx2);
          float ih = iy2 - iy1; ih = ih > 0.0f ? ih : 0.0f;
          float iw = ix2 - ix1; iw = iw > 0.0f ? iw : 0.0f;
          float inter = ih * iw;
          float uni   = s_area[j] + ba - inter;
          float iou   = inter / fmaxf(uni, 1e-8f);
          if (iou > IOU_THR) s_active[j] = 0;   // suppresses `sel` itself too (IoU=1)
        }
      }
    }
    // no barrier needed: each thread only touches its own s_active slots;
    // next-step LDS publishes are fenced by B1/B2.
  }
}

extern "C" void kernel_launch(void* const* d_in, const int* in_sizes, int n_in,
                              void* d_out, int out_size, void* d_ws, size_t ws_size,
                              hipStream_t stream) {
  (void)n_in; (void)out_size; (void)d_ws; (void)ws_size;
  const int*   anchor_idxes = (const int*)d_in[0];   // [N,2] int32
  const float* anchors      = (const float*)d_in[1]; // [N,4] float32
  const int*   class_ids    = (const int*)d_in[2];   // [N] int32
  const float* scores       = (const float*)d_in[3]; // [N] float32
  float*       out          = (float*)d_out;
  const int n = in_sizes[2];

  NMS_FPN_56427280335586_kernel<<<dim3(NUM_CLASSES), dim3(BLOCK), 0, stream>>>(
      anchor_idxes, anchors, class_ids, scores, out, n);
}